// EnhancedKeypointLoss_38457137168611
// MI455X (gfx1250) — compile-verified
//
#include <hip/hip_runtime.h>
#include <math.h>

typedef __attribute__((ext_vector_type(2))) float v2f;
typedef __attribute__((ext_vector_type(8))) float v8f;

#define KPT 17
#define WPB 8            // waves per block (256 threads, wave32)
#define TPB (WPB * 32)
#define MAXBLK 1024

__device__ __forceinline__ float wred32(float v) {
#pragma unroll
  for (int off = 16; off > 0; off >>= 1) v += __shfl_xor(v, off, 32);
  return v;  // butterfly: result broadcast to all lanes
}

__device__ __forceinline__ float safe_sqrtf(float sq) {
  return sq > 0.0f ? sqrtf(sq) : 0.0f;
}

__device__ __forceinline__ float smooth_l1(float x) {
  float ax = fabsf(x);
  return ax < 1.0f ? 0.5f * x * x : ax - 0.5f;
}

__device__ __forceinline__ float clamp1(float x) {
  return fminf(fmaxf(x, -1.0f), 1.0f);
}

struct WaveSh {
  float px[KPT], py[KPT], gx[KPT], gy[KPT];
  float vm[KPT], mraw[KPT];
  float nP[KPT], nG[KPT];
};

__global__ void __launch_bounds__(TPB) kpl_partial(
    const float* __restrict__ pred, const float* __restrict__ gt,
    const float* __restrict__ mask, const float* __restrict__ sigmas,
    float* __restrict__ partials, int N) {
  __shared__ WaveSh sh[WPB];
  __shared__ float inv8s2[KPT];
  __shared__ float wpart[WPB][5];

  const int tid = threadIdx.x;
  const int wave = tid >> 5;
  const int lane = tid & 31;
  const int kc = lane & 15;          // C/D tile column for this lane
  const int hi = (lane >> 4) << 3;   // tile row base: 0 or 8

  if (tid < KPT) {
    float s = sigmas[tid];
    inv8s2[tid] = 1.0f / (8.0f * s * s);
  }

  float acc_base = 0.0f, acc_rl = 0.0f, acc_valid = 0.0f;
  float acc_ang = 0.0f, acc_cnt = 0.0f;

  for (int b0 = blockIdx.x * WPB; b0 < N; b0 += gridDim.x * WPB) {
    const int b = b0 + wave;
    const bool active = (b < N);
    WaveSh& S = sh[wave];

    __syncthreads();
    if (lane < KPT) {
      float pxv = 0, pyv = 0, gxv = 0, gyv = 0, mv = 0;
      if (active) {
        const float* pp = pred + ((size_t)b * KPT + lane) * 3;
        const float* gg = gt + ((size_t)b * KPT + lane) * 3;
        pxv = pp[0]; pyv = pp[1];
        gxv = gg[0]; gyv = gg[1];
        mv = mask[(size_t)b * KPT + lane];
      }
      S.px[lane] = pxv; S.py[lane] = pyv;
      S.gx[lane] = gxv; S.gy[lane] = gyv;
      S.mraw[lane] = mv;
      S.vm[lane] = (mv > 0.0f) ? 1.0f : 0.0f;
    }
    __syncthreads();

    // ---------------- base (OKS) term ----------------
    float t = 0.0f, nv = 0.0f;
    if (lane < KPT) {
      float dx = S.px[lane] - S.gx[lane];
      float dy = S.py[lane] - S.gy[lane];
      float d = dx * dx + dy * dy;
      float e = d * inv8s2[lane];
      t = (1.0f - expf(-e)) * S.mraw[lane];
      nv = (S.mraw[lane] != 0.0f) ? 1.0f : 0.0f;
    }
    float sum_t = wred32(t);
    float sum_nv = wred32(nv);
    acc_base += ((float)KPT / (sum_nv + 1e-9f)) * sum_t;

    // per-sample mask values cached in registers (branchless selects later)
    float vmi[8];
#pragma unroll
    for (int r = 0; r < 8; ++r) vmi[r] = S.vm[r + hi];
    const float vmk = S.vm[kc];
    const float vm16 = S.vm[16];
    const float vml = (lane < KPT) ? S.vm[lane] : 0.0f;

    // ---------------- ratio term ----------------
    float pdv[10], gdv[10], pmv[10];
    float s_np = 0.0f, s_pd = 0.0f, s_gd = 0.0f;
#pragma unroll
    for (int it = 0; it < 10; ++it) {
      int e = lane + 32 * it;
      float pd = 0, gd = 0, pm = 0;
      if (e < KPT * KPT) {
        int i = e / KPT, k = e - i * KPT;
        float dx = S.px[i] - S.px[k], dy = S.py[i] - S.py[k];
        pd = safe_sqrtf(dx * dx + dy * dy);
        dx = S.gx[i] - S.gx[k]; dy = S.gy[i] - S.gy[k];
        gd = safe_sqrtf(dx * dx + dy * dy);
        float vv = S.vm[i] * S.vm[k];
        pm = (k > i) ? vv : 0.0f;
      }
      pdv[it] = pd; gdv[it] = gd; pmv[it] = pm;
      s_np += pm; s_pd += pd * pm; s_gd += gd * pm;
    }
    s_np = wred32(s_np); s_pd = wred32(s_pd); s_gd = wred32(s_gd);
    float denomp = fmaxf(s_np, 1.0f);
    float ipm = 1.0f / (s_pd / denomp + 1e-6f);
    float igm = 1.0f / (s_gd / denomp + 1e-6f);
    float s_rl = 0.0f;
#pragma unroll
    for (int it = 0; it < 10; ++it)
      s_rl += smooth_l1(pdv[it] * ipm - gdv[it] * igm) * pmv[it];
    s_rl = wred32(s_rl);
    float valid = (s_np >= 1.0f) ? 1.0f : 0.0f;
    acc_rl += (s_rl / denomp) * valid;
    acc_valid += valid;

    // ---------------- angle term ----------------
    float l_ang = 0.0f, l_cnt = 0.0f;
    for (int j = 0; j < KPT; ++j) {
      __syncthreads();
      float dpx = 0, dpy = 0, dgx = 0, dgy = 0;
      float nPown = 0, nGown = 0;
      if (lane < KPT) {
        dpx = S.px[lane] - S.px[j]; dpy = S.py[lane] - S.py[j];
        dgx = S.gx[lane] - S.gx[j]; dgy = S.gy[lane] - S.gy[j];
        nPown = safe_sqrtf(dpx * dpx + dpy * dpy);
        nGown = safe_sqrtf(dgx * dgx + dgy * dgy);
        S.nP[lane] = nPown;
        S.nG[lane] = nGown;
      }
      __syncthreads();

      // 16x16 block of D·D^T via V_WMMA_F32_16X16X4_F32.
      // A layout: VGPR0 = {lanes 0-15: K=0 (=Dx), lanes 16-31: K=2 (=0 pad)},
      //           VGPR1 = {K=1 (=Dy), K=3 (=0 pad)}; B assumed mirrored, so
      //           the same register pair serves as both operands (symmetric Gram).
      v2f aP, aG;
      aP.x = (lane < 16) ? dpx : 0.0f; aP.y = (lane < 16) ? dpy : 0.0f;
      aG.x = (lane < 16) ? dgx : 0.0f; aG.y = (lane < 16) ? dgy : 0.0f;
      v8f cP = {}; v8f cG = {};
      cP = __builtin_amdgcn_wmma_f32_16x16x4_f32(false, aP, false, aP,
                                                 (short)0, cP, false, false);
      cG = __builtin_amdgcn_wmma_f32_16x16x4_f32(false, aG, false, aG,
                                                 (short)0, cG, false, false);

      // hoisted LDS reads: norms for the lane's 8 tile rows + its column
      float nPi[8], nGi[8];
#pragma unroll
      for (int r = 0; r < 8; ++r) { nPi[r] = S.nP[r + hi]; nGi[r] = S.nG[r + hi]; }
      const float nPk = S.nP[kc], nGk = S.nG[kc];
      const float vmj = S.vm[j];

#pragma unroll
      for (int r = 0; r < 8; ++r) {
        int i = r + hi;
        float trip = ((i != j) && (kc != j) && (i != kc)) ? 1.0f : 0.0f;
        float m = trip * vmi[r] * vmk * vmj;
        float cp = clamp1(cP[r] / (nPi[r] * nPk + 1e-6f));
        float cg = clamp1(cG[r] / (nGi[r] * nGk + 1e-6f));
        float dd = acosf(cp) - acosf(cg);
        l_ang += dd * dd * m;
        l_cnt += m;
      }

      // Border (keypoint 16): ap[16,k]==ap[k,16] and mask is i<->k symmetric,
      // so lane l<16 computes entry (16,l) once, weighted 2x; (16,16) masked.
      // Lane 16's D-vector and norms broadcast via shuffle; rest from own regs.
      {
        float dpx16 = __shfl(dpx, 16, 32), dpy16 = __shfl(dpy, 16, 32);
        float dgx16 = __shfl(dgx, 16, 32), dgy16 = __shfl(dgy, 16, 32);
        float nP16 = __shfl(nPown, 16, 32), nG16 = __shfl(nGown, 16, 32);
        float bsel = ((lane < 16) && (lane != j) && (j != 16)) ? 2.0f : 0.0f;
        float m = bsel * vm16 * vml * vmj;
        float dotP = dpx16 * dpx + dpy16 * dpy;
        float dotG = dgx16 * dgx + dgy16 * dgy;
        float cp = clamp1(dotP / (nP16 * nPown + 1e-6f));
        float cg = clamp1(dotG / (nG16 * nGown + 1e-6f));
        float dd = acosf(cp) - acosf(cg);
        l_ang += dd * dd * m;
        l_cnt += m;
      }
    }
    acc_ang += wred32(l_ang);
    acc_cnt += wred32(l_cnt);
  }

  if (lane == 0) {
    wpart[wave][0] = acc_base;
    wpart[wave][1] = acc_rl;
    wpart[wave][2] = acc_valid;
    wpart[wave][3] = acc_ang;
    wpart[wave][4] = acc_cnt;
  }
  __syncthreads();
  if (tid < 5) {
    float s = 0.0f;
    for (int w = 0; w < WPB; ++w) s += wpart[w][tid];
    partials[(size_t)blockIdx.x * 5 + tid] = s;
  }
}

__global__ void __launch_bounds__(256) kpl_final(const float* __restrict__ partials,
                                                 float* __restrict__ out,
                                                 int nblk, int N) {
  __shared__ float red[256][5];
  float loc[5] = {0, 0, 0, 0, 0};
  for (int i = threadIdx.x; i < nblk; i += 256) {
#pragma unroll
    for (int q = 0; q < 5; ++q) loc[q] += partials[(size_t)i * 5 + q];
  }
#pragma unroll
  for (int q = 0; q < 5; ++q) red[threadIdx.x][q] = loc[q];
  __syncthreads();
  for (int s = 128; s > 0; s >>= 1) {
    if (threadIdx.x < s) {
#pragma unroll
      for (int q = 0; q < 5; ++q) red[threadIdx.x][q] += red[threadIdx.x + s][q];
    }
    __syncthreads();
  }
  if (threadIdx.x == 0) {
    float base = red[0][0] / ((float)N * (float)KPT);
    float ratio = red[0][1] / fmaxf(red[0][2], 1.0f);
    float angle = red[0][3] / fmaxf(red[0][4], 1.0f);
    out[0] = 1.0f * base + 0.2f * ratio + 0.2f * angle;
  }
}

extern "C" void kernel_launch(void* const* d_in, const int* in_sizes, int n_in,
                              void* d_out, int out_size, void* d_ws, size_t ws_size,
                              hipStream_t stream) {
  const float* pred = (const float*)d_in[0];
  const float* gt = (const float*)d_in[1];
  const float* mask = (const float*)d_in[2];
  const float* sig = (const float*)d_in[3];
  int N = in_sizes[0] / (KPT * 3);

  float* partials = (float*)d_ws;  // nblk*5 floats (<= 20 KB)
  int nblk = (N + WPB - 1) / WPB;
  if (nblk > MAXBLK) nblk = MAXBLK;

  kpl_partial<<<nblk, TPB, 0, stream>>>(pred, gt, mask, sig, partials, N);
  kpl_final<<<1, 256, 0, stream>>>(partials, (float*)d_out, nblk, N);
}